// VoxelSetAbstraction_20469814133332
// MI455X (gfx1250) — compile-verified
//
#include <hip/hip_runtime.h>
#include <math.h>

// CDNA5 / gfx1250. wave32.
typedef float v2f __attribute__((ext_vector_type(2)));
typedef float v4f __attribute__((ext_vector_type(4)));
typedef float v8f __attribute__((ext_vector_type(8)));

#define PC_LO_X 0.0f
#define PC_LO_Y (-40.0f)
#define INV_VOX 2.5f            // 1 / (0.05 * 8)
#define RADIUS  2.4f
#define HGT 188
#define WID 188
#define NCH 256

// ---------------------------------------------------------------------------
// Kernel 1: per-ROI prep. Packs the WMMA B fragment (K x N = 4 x 16 f32 tile)
// in the ISA VGPR layout (v0: lanes0-15=K0, lanes16-31=K2; v1: K1/K3), and
// the squared radius threshold (0.5*||dims|| + 2.4)^2.
// dist^2(p, r) = |p|^2 + [ (x,y,z,1) . (-2rx, -2ry, -2rz, |r|^2) ]
// ---------------------------------------------------------------------------
__global__ void vsa_prep(const float* __restrict__ rois,
                         float* __restrict__ Bmat,   // [M/16][64]
                         float* __restrict__ thr2,   // [M]
                         int M) {
  int m = blockIdx.x * blockDim.x + threadIdx.x;
  if (m >= M) return;
  float rx = rois[m * 7 + 0], ry = rois[m * 7 + 1], rz = rois[m * 7 + 2];
  float dx = rois[m * 7 + 3], dy = rois[m * 7 + 4], dz = rois[m * 7 + 5];
  int t = m >> 4, col = m & 15;
  float* Bt = Bmat + t * 64;
  Bt[col]      = -2.0f * rx;                 // v0, lanes 0-15  : K=0
  Bt[16 + col] = -2.0f * rz;                 // v0, lanes 16-31 : K=2
  Bt[32 + col] = -2.0f * ry;                 // v1, lanes 0-15  : K=1
  Bt[48 + col] = rx * rx + ry * ry + rz * rz;// v1, lanes 16-31 : K=3
  float th = 0.5f * sqrtf(dx * dx + dy * dy + dz * dz) + RADIUS;
  thr2[m] = th * th;
}

// ---------------------------------------------------------------------------
// Kernel 2: mask via WMMA. One wave handles 16 points; 8 tiles of 16 ROIs each
// -> 8 x v_wmma_f32_16x16x4_f32. D layout: VGPR k, lanes 0-15 -> row M=k,
// lanes 16-31 -> row M=8+k, col N = lane&15. Running (min, thr-of-min) per
// slot across tiles, then shfl_xor min-reduction across the 16 columns.
// Control flow is wave-uniform before the WMMAs -> EXEC all ones.
// ---------------------------------------------------------------------------
__global__ void vsa_mask(const float* __restrict__ points,
                         const float* __restrict__ Bmat,
                         const float* __restrict__ thr2,
                         float* __restrict__ mask,
                         int N, int ntile) {
  const int lane  = threadIdx.x & 31;
  const int wave  = threadIdx.x >> 5;
  const int group = blockIdx.x * (blockDim.x >> 5) + wave;
  const int ngroups = N >> 4;
  if (group >= ngroups) return;            // wave-uniform guard
  const int l16   = lane & 15;
  const int halfi = lane >> 4;
  const int pidx  = group * 16 + l16;

  // A fragment (16x4 f32): lanes0-15 {v0=x, v1=y}; lanes16-31 {v0=z, v1=1}
  float x_or_z = points[pidx * 3 + 2 * halfi];
  float yv     = points[pidx * 3 + 1];
  v2f a;
  a.x = x_or_z;
  a.y = halfi ? 1.0f : yv;

  // |p|^2: half0 holds x^2+y^2, half1 holds z^2; combine across halves.
  float p2part = x_or_z * x_or_z + (halfi ? 0.0f : yv * yv);
  float p2 = p2part + __shfl_xor(p2part, 16);   // lane l: |p_{base+(l&15)}|^2

  float minv[8], thrm[8];
#pragma unroll
  for (int k = 0; k < 8; ++k) { minv[k] = 3.4e38f; thrm[k] = 0.0f; }

  for (int t = 0; t < ntile; ++t) {
    v2f b;
    b.x = Bmat[t * 64 + lane];
    b.y = Bmat[t * 64 + 32 + lane];
    v8f c = {};
    v8f d = __builtin_amdgcn_wmma_f32_16x16x4_f32(
        false, a, false, b, (short)0, c, false, false);
    float th = thr2[t * 16 + l16];
#pragma unroll
    for (int k = 0; k < 8; ++k) {
      if (d[k] < minv[k]) { minv[k] = d[k]; thrm[k] = th; }
    }
  }

  // Min-reduce across the 16 lanes of each half, carrying the threshold.
#pragma unroll
  for (int mlk = 1; mlk < 16; mlk <<= 1) {
#pragma unroll
    for (int k = 0; k < 8; ++k) {
      float ov = __shfl_xor(minv[k], mlk);
      float ot = __shfl_xor(thrm[k], mlk);
      if (ov < minv[k]) { minv[k] = ov; thrm[k] = ot; }
    }
  }

  // q = min dist^2 = min(-2 p.r + |r|^2) + |p|^2 (per slot's point).
  float q[8];
#pragma unroll
  for (int k = 0; k < 8; ++k)
    q[k] = minv[k] + __shfl(p2, k + 8 * halfi);

  if (l16 < 8) {
    int k = l16;
    mask[group * 16 + k + 8 * halfi] = (q[k] < thrm[k]) ? 1.0f : 0.0f;
  }
}

// ---------------------------------------------------------------------------
// Kernel 3: bilinear gather + masked write. 64 threads per point (4 ch each).
// BEV image (36 MB) is L2-resident; output (205 MB) streamed with NT stores.
// ---------------------------------------------------------------------------
__global__ void vsa_feat(const float* __restrict__ points,
                         const float* __restrict__ bev,
                         const float* __restrict__ mask,
                         float* __restrict__ out,
                         int N) {
  long tid = (long)blockIdx.x * blockDim.x + threadIdx.x;
  int p  = (int)(tid >> 6);
  int cb = ((int)tid & 63) * 4;
  if (p >= N) return;

  float px = points[p * 3 + 0];
  float py = points[p * 3 + 1];
  float x = (px - PC_LO_X) * INV_VOX;
  float y = (py - PC_LO_Y) * INV_VOX;
  int xf = (int)floorf(x), yf = (int)floorf(y);
  int x0 = min(max(xf, 0), WID - 1);
  int x1 = min(max(xf + 1, 0), WID - 1);
  int y0 = min(max(yf, 0), HGT - 1);
  int y1 = min(max(yf + 1, 0), HGT - 1);
  float x0f = (float)x0, x1f = (float)x1, y0f = (float)y0, y1f = (float)y1;
  float msk = mask[p];
  float wa = (x1f - x) * (y1f - y) * msk;
  float wb = (x1f - x) * (y - y0f) * msk;
  float wc = (x - x0f) * (y1f - y) * msk;
  float wd = (x - x0f) * (y - y0f) * msk;

  const v4f* Ia = (const v4f*)(bev + ((long)(y0 * WID + x0) * NCH + cb));
  const v4f* Ib = (const v4f*)(bev + ((long)(y1 * WID + x0) * NCH + cb));
  const v4f* Ic = (const v4f*)(bev + ((long)(y0 * WID + x1) * NCH + cb));
  const v4f* Id = (const v4f*)(bev + ((long)(y1 * WID + x1) * NCH + cb));

  v4f r = Ia[0] * wa + Ib[0] * wb + Ic[0] * wc + Id[0] * wd;
  __builtin_nontemporal_store(r, (v4f*)(out + (long)p * NCH + cb));
}

// ---------------------------------------------------------------------------
extern "C" void kernel_launch(void* const* d_in, const int* in_sizes, int n_in,
                              void* d_out, int out_size, void* d_ws, size_t ws_size,
                              hipStream_t stream) {
  const float* points = (const float*)d_in[0];   // (N, 3)
  const float* rois   = (const float*)d_in[1];   // (M, 7)
  const float* bev    = (const float*)d_in[2];   // (188, 188, 256)
  float* out = (float*)d_out;

  int N = in_sizes[0] / 3;   // 200000
  int M = in_sizes[1] / 7;   // 128
  int ntile = M / 16;        // 8

  // Workspace layout: mask[N] | Bmat[ntile*64] | thr2[M]  (~803 KB)
  float* ws   = (float*)d_ws;
  float* mask = ws;
  float* Bmat = ws + N;
  float* thr2 = Bmat + ntile * 64;

  vsa_prep<<<(M + 127) / 128, 128, 0, stream>>>(rois, Bmat, thr2, M);

  int ngroups = N / 16;                       // 12500 (N divisible by 16)
  int wavesPerBlock = 8;                      // 256 threads
  int mblocks = (ngroups + wavesPerBlock - 1) / wavesPerBlock;
  vsa_mask<<<mblocks, 256, 0, stream>>>(points, Bmat, thr2, mask, N, ntile);

  long total = (long)N * 64;
  int fblocks = (int)((total + 255) / 256);   // 50000
  vsa_feat<<<fblocks, 256, 0, stream>>>(points, bev, mask, out, N);
}